// ReToroidalization_57921928953945
// MI455X (gfx1250) — compile-verified
//
#include <hip/hip_runtime.h>

// ReToroidalization for MI455X (gfx1250, wave32).
// Memory-bound (8.3 GFLOP vs 512 MB traffic -> ~22us floor at 23.3 TB/s),
// so: stream x once (nontemporal), stream out once, weights stay hot in L2
// (1 MB << 192 MB), and do the per-block 31x32 projections with
// V_WMMA_F32_16X16X4_F32 (full f32 precision to match the f32 reference).
//
// Pad-row trick: rank row 31 of the C tile is never read back (output elem 31
// comes from sigma*sin), so the A-row address for the pad lane is clamped to
// row 30 instead of branching on EXEC to zero it.
// Reciprocal: den = 1 + phi^2 >= 1, so v_rcp_f32 + one Newton step (~0.5 ulp)
// replaces the ~10-op IEEE divide chain -> big VALU cut in the scan phase.

#define DIM   4096
#define BLK   32
#define NBLK  128
#define RANK  31
#define TM    32      // tokens per wave (and per workgroup)
#define WAVES 4       // blocks (n) per workgroup, one per wave
#define LSTR  36      // LDS row stride in words: 8*36 mod 64 = 32 -> the two
                      // lane-half C writes land on disjoint bank groups

typedef __attribute__((ext_vector_type(2))) float v2f;
typedef __attribute__((ext_vector_type(4))) float v4f;
typedef __attribute__((ext_vector_type(8))) float v8f;

__global__ __launch_bounds__(32 * WAVES)
void retor_kernel(const float* __restrict__ x,
                  const float* __restrict__ Wr,
                  const float* __restrict__ Wt,
                  const float* __restrict__ gain,
                  float* __restrict__ out)
{
    __shared__ float lds[WAVES][2][32 * LSTR];   // [wave][rho|phi][rank][token]

    const int wave = threadIdx.x >> 5;
    const int lane = threadIdx.x & 31;
    const int n    = blockIdx.y * WAVES + wave;  // block column
    const long t0  = (long)blockIdx.x * TM;      // first token of tile

    const int h  = lane >> 4;    // K-pair half (A/B VGPR0 holds K=2h, VGPR1 K=2h+1)
    const int ml = lane & 15;    // row (A) / col (B) within 16-wide tile

    float* ldsR = &lds[wave][0][0];
    float* ldsP = &lds[wave][1][0];

    const float* Wrn = Wr + (size_t)n * RANK * BLK;
    const float* Wtn = Wt + (size_t)n * RANK * BLK;

    const int r0 = ml;                                    // rank tile 0: rows 0..15
    const int r1 = (16 + ml < RANK) ? 16 + ml : RANK - 1; // tile 1, pad lane clamped

    // ---- Phase 1: rho = W_rho * x^T, phi = W_theta * x^T  (f32 WMMA) ----
    for (int nt = 0; nt < 2; ++nt) {             // two 16-token N tiles
        v8f accR0 = {}, accR1 = {}, accP0 = {}, accP1 = {};
        const long tok    = t0 + nt * 16 + ml;   // B: lane ml supplies token col
        const float* xrow = x + (size_t)tok * DIM + (size_t)n * BLK;

#pragma unroll
        for (int kc = 0; kc < 8; ++kc) {         // K = 32 in chunks of 4
            const int kcol = kc * 4 + h * 2;
            // B fragment: x^T, two consecutive K values of this lane's token.
            // Streamed once -> nontemporal, keep near caches for weights.
            v2f b = __builtin_nontemporal_load((const v2f*)(xrow + kcol));
            // A fragments: weight rows, same K-pair striping (L2-hot)
            v2f aR0 = *(const v2f*)(Wrn + r0 * BLK + kcol);
            v2f aT0 = *(const v2f*)(Wtn + r0 * BLK + kcol);
            v2f aR1 = *(const v2f*)(Wrn + r1 * BLK + kcol);
            v2f aT1 = *(const v2f*)(Wtn + r1 * BLK + kcol);
            accR0 = __builtin_amdgcn_wmma_f32_16x16x4_f32(false, aR0, false, b,
                                                          (short)0, accR0, false, false);
            accR1 = __builtin_amdgcn_wmma_f32_16x16x4_f32(false, aR1, false, b,
                                                          (short)0, accR1, false, false);
            accP0 = __builtin_amdgcn_wmma_f32_16x16x4_f32(false, aT0, false, b,
                                                          (short)0, accP0, false, false);
            accP1 = __builtin_amdgcn_wmma_f32_16x16x4_f32(false, aT1, false, b,
                                                          (short)0, accP1, false, false);
        }

        // C layout: VGPR i -> M = i (lanes 0-15) / i+8 (lanes 16-31), N = ml.
        const int colw = nt * 16 + ml;
#pragma unroll
        for (int i = 0; i < 8; ++i) {
            const int rk0 = i + 8 * h;        // rank tile 0
            const int rk1 = 16 + i + 8 * h;   // rank tile 1 (row 31 = dead pad)
            ldsR[rk0 * LSTR + colw] = accR0[i];
            ldsR[rk1 * LSTR + colw] = accR1[i];
            ldsP[rk0 * LSTR + colw] = accP0[i];
            ldsP[rk1 * LSTR + colw] = accP1[i];
        }
    }
    __syncthreads();

    // ---- Phase 2: per-token Weierstrass trig + 31-step rank scan ----
    // lane == token within the tile; LDS reads are stride-1 across lanes.
    const float* gn = gain + (size_t)n * (RANK - 1);
    float garr[RANK - 1];
#pragma unroll
    for (int j = 0; j < RANK - 1; ++j) garr[j] = gn[j];   // vectorizes to b128 loads

    float outv[BLK];
    float sigma = 0.0f, prev_sin = 0.0f;

#pragma unroll
    for (int r = 0; r < RANK; ++r) {
        float rho = ldsR[r * LSTR + lane];
        float phi = ldsP[r * LSTR + lane];
        float p2  = phi * phi;
        float den = 1.0f + p2;                 // >= 1: rcp is safe
        float inv = __builtin_amdgcn_rcpf(den);
        inv = inv * (2.0f - den * inv);        // one Newton step -> ~0.5 ulp
        float ct  = (1.0f - p2) * inv;
        float st  = (2.0f * phi) * inv;
        if (r == 0) sigma = rho;
        else        sigma = sigma * (garr[r - 1] * prev_sin) + rho;
        outv[r]  = sigma * ct;
        prev_sin = st;
    }
    outv[BLK - 1] = sigma * prev_sin;   // sigma_30 * sin_t_30

    float* orow = out + (size_t)(t0 + lane) * DIM + (size_t)n * BLK;
#pragma unroll
    for (int j = 0; j < 8; ++j) {
        v4f v;
        v[0] = outv[4 * j + 0];
        v[1] = outv[4 * j + 1];
        v[2] = outv[4 * j + 2];
        v[3] = outv[4 * j + 3];
        *(v4f*)(orow + 4 * j) = v;
    }
}

extern "C" void kernel_launch(void* const* d_in, const int* in_sizes, int n_in,
                              void* d_out, int out_size, void* d_ws, size_t ws_size,
                              hipStream_t stream) {
    const float* x    = (const float*)d_in[0];
    const float* Wr   = (const float*)d_in[1];
    const float* Wt   = (const float*)d_in[2];
    const float* gain = (const float*)d_in[3];
    float* out        = (float*)d_out;

    const long tokens = (long)in_sizes[0] / DIM;   // 16384 for (4,4096,4096)
    dim3 grid((unsigned)((tokens + TM - 1) / TM), NBLK / WAVES);
    dim3 block(32 * WAVES);
    retor_kernel<<<grid, block, 0, stream>>>(x, Wr, Wt, gain, out);
}